// SeriesDecomp_27788438405464
// MI455X (gfx1250) — compile-verified
//
#include <hip/hip_runtime.h>

// Series decomposition (Autoformer-style): moving average K=2049 with
// replicate padding over rows of L=2048, x_s = x - x_t.
// Memory-bound: 128 MiB in + 256 MiB out = 384 MiB -> ~17us floor @ 23.3 TB/s.
// CDNA5 path: async global->LDS row staging (ASYNCcnt + s_wait_asynccnt),
// wave32 shuffle scan (no barriers), non-temporal b128 output stores.

#define ROW_L   2048
#define TPB     256
#define NWAVES  (TPB / 32)
#define EPT     8          // elements per thread (2 x float4)
#define HALF_W  1024       // p = ceil((L-1)/2)
#define INV_K   (1.0f / 2049.0f)

typedef int   v4i __attribute__((ext_vector_type(4)));
typedef float v4f __attribute__((ext_vector_type(4)));
typedef __attribute__((address_space(1))) v4i* gbl_v4i_ptr;
typedef __attribute__((address_space(3))) v4i* lds_v4i_ptr;
typedef __attribute__((address_space(1))) void* gbl_void_ptr;
typedef __attribute__((address_space(3))) void* lds_void_ptr;

// ---- CDNA5 async global->LDS copy (B128) -----------------------------------
__device__ __forceinline__ void async_copy_b128_to_lds(const float* g, float* l) {
#if __has_builtin(__builtin_amdgcn_global_load_async_to_lds_b128)
    __builtin_amdgcn_global_load_async_to_lds_b128(
        (gbl_v4i_ptr)(gbl_void_ptr)(void*)g,
        (lds_v4i_ptr)(lds_void_ptr)(void*)l,
        /*imm offset*/ 0, /*cpol*/ 0);
#else
    unsigned lds_addr =
        (unsigned)(unsigned long long)(__attribute__((address_space(3))) float*)l;
    unsigned long long gaddr = (unsigned long long)g;
    asm volatile("global_load_async_to_lds_b128 %0, %1, off"
                 :: "v"(lds_addr), "v"(gaddr)
                 : "memory");
#endif
}

__device__ __forceinline__ void wait_async_zero() {
#if __has_builtin(__builtin_amdgcn_s_wait_asynccnt)
    __builtin_amdgcn_s_wait_asynccnt(0);
#else
    asm volatile("s_wait_asynccnt 0" ::: "memory");
#endif
}

// ---- kernel ----------------------------------------------------------------
__global__ __launch_bounds__(TPB) void series_decomp_kernel(
        const float* __restrict__ x,
        float* __restrict__ xs_out,
        float* __restrict__ xt_out) {
    __shared__ float sx[ROW_L];       // raw row
    __shared__ float sp[ROW_L];       // inclusive prefix sums
    __shared__ float swtot[NWAVES];   // per-wave totals

    const int t    = threadIdx.x;
    const int lane = t & 31;
    const int wave = t >> 5;
    const long long row = blockIdx.x;
    const float* gin = x + row * (long long)ROW_L;

    // Stage the 8KB row into LDS with two async b128 copies per thread.
    async_copy_b128_to_lds(gin + t * 4,          &sx[t * 4]);
    async_copy_b128_to_lds(gin + HALF_W + t * 4, &sx[HALF_W + t * 4]);
    wait_async_zero();          // our wave's async copies landed in LDS
    __syncthreads();            // ...and everyone else's

    // Per-thread serial inclusive scan over 8 contiguous elements.
    float a[EPT];
    const int i0 = t * EPT;
    v4f v0 = *(const v4f*)&sx[i0];
    v4f v1 = *(const v4f*)&sx[i0 + 4];
    a[0] = v0.x; a[1] = v0.y; a[2] = v0.z; a[3] = v0.w;
    a[4] = v1.x; a[5] = v1.y; a[6] = v1.z; a[7] = v1.w;

    float pref[EPT];
    float run = 0.0f;
#pragma unroll
    for (int j = 0; j < EPT; ++j) { run += a[j]; pref[j] = run; }

    // Wave32 inclusive shuffle scan of per-thread totals (no barriers, no LDS).
    float inc = run;
#pragma unroll
    for (int d = 1; d < 32; d <<= 1) {
        float n = __shfl_up(inc, d, 32);
        if (lane >= d) inc += n;
    }

    // One LDS exchange for the 8 wave totals.
    if (lane == 31) swtot[wave] = inc;
    __syncthreads();
    float woff = 0.0f;
#pragma unroll
    for (int w = 0; w < NWAVES; ++w)
        woff += (w < wave) ? swtot[w] : 0.0f;

    const float excl = woff + inc - run;   // exclusive offset for this chunk

    // Publish the row-wide inclusive prefix.
    v4f p0 = { excl + pref[0], excl + pref[1], excl + pref[2], excl + pref[3] };
    v4f p1 = { excl + pref[4], excl + pref[5], excl + pref[6], excl + pref[7] };
    *(v4f*)&sp[i0]     = p0;
    *(v4f*)&sp[i0 + 4] = p1;
    __syncthreads();

    const float x0 = sx[0];
    const float xN = sx[ROW_L - 1];

    // Window [i-1024, i+1024] clipped to [0,2047]; out-of-range entries are
    // replicated edge samples. Since K=2049 > L, exactly one edge is active.
    float xt[EPT];
#pragma unroll
    for (int j = 0; j < EPT; ++j) {
        const int i  = i0 + j;
        const int hi = i + HALF_W;
        const int lo = i - HALF_W;
        const int hic = (hi < ROW_L) ? hi : (ROW_L - 1);
        const float mid  = sp[hic] - ((lo > 0) ? sp[lo - 1] : 0.0f);
        const float edge = (i < HALF_W) ? (float)(HALF_W - i) * x0
                                        : (float)(i - (HALF_W - 1)) * xN;
        xt[j] = (mid + edge) * INV_K;
    }

    // Streamed output: written once, never re-read -> non-temporal b128 stores
    // (keep the 192MB L2 for input lines instead).
    const long long base = row * (long long)ROW_L + i0;
    v4f s0 = { a[0] - xt[0], a[1] - xt[1], a[2] - xt[2], a[3] - xt[3] };
    v4f s1 = { a[4] - xt[4], a[5] - xt[5], a[6] - xt[6], a[7] - xt[7] };
    v4f t0 = { xt[0], xt[1], xt[2], xt[3] };
    v4f t1 = { xt[4], xt[5], xt[6], xt[7] };
    __builtin_nontemporal_store(s0, (v4f*)&xs_out[base]);
    __builtin_nontemporal_store(s1, (v4f*)&xs_out[base + 4]);
    __builtin_nontemporal_store(t0, (v4f*)&xt_out[base]);
    __builtin_nontemporal_store(t1, (v4f*)&xt_out[base + 4]);
}

// ---- launcher --------------------------------------------------------------
extern "C" void kernel_launch(void* const* d_in, const int* in_sizes, int n_in,
                              void* d_out, int out_size, void* d_ws, size_t ws_size,
                              hipStream_t stream) {
    (void)n_in; (void)d_ws; (void)ws_size; (void)out_size;
    const float* x = (const float*)d_in[0];
    float* out = (float*)d_out;

    const int rows = in_sizes[0] / ROW_L;          // 32*512 = 16384
    float* xs = out;                               // first tuple element
    float* xt = out + (long long)rows * ROW_L;     // second tuple element

    series_decomp_kernel<<<rows, TPB, 0, stream>>>(x, xs, xt);
}